// Net2_55473797595450
// MI455X (gfx1250) — compile-verified
//
#include <hip/hip_runtime.h>
#include <hip/hip_bf16.h>

#define N_NODES 50000
#define N_EDGES 800000
#define N_GRAPHS 512
#define C 128
#define NUM_EMB 119

typedef __attribute__((ext_vector_type(2))) float v2f;
typedef __attribute__((ext_vector_type(8))) float v8f;

// ---------------- graph normalization -----------------------------------

__global__ void deg_kernel(const int* __restrict__ row, float* __restrict__ deg, int E) {
    int e = blockIdx.x * blockDim.x + threadIdx.x;
    if (e < E) atomicAdd(&deg[row[e]], 1.0f);
}

__global__ void dis_kernel(const float* __restrict__ deg, float* __restrict__ dis, int N) {
    int i = blockIdx.x * blockDim.x + threadIdx.x;
    if (i < N) {
        float d = deg[i];
        dis[i] = (d > 0.0f) ? rsqrtf(fmaxf(d, 1.0f)) : 0.0f;
    }
}

__global__ void norm_kernel(const int* __restrict__ row, const int* __restrict__ col,
                            const float* __restrict__ dis, float* __restrict__ norm, int E) {
    int e = blockIdx.x * blockDim.x + threadIdx.x;
    if (e < E) norm[e] = -(dis[row[e]] * dis[col[e]]);
}

// ---------------- atom encoder -------------------------------------------

__global__ void atom_encoder_kernel(const float* __restrict__ emb, const int* __restrict__ x,
                                    float* __restrict__ h) {
    int n = blockIdx.x;
    int c = threadIdx.x;
    float s = 0.0f;
    #pragma unroll
    for (int f = 0; f < 9; ++f) {
        int idx = x[n * 9 + f];
        s += emb[(f * NUM_EMB + idx) * C + c];
    }
    h[n * C + c] = s;
}

// ---------------- sparse Lhat scatter-add --------------------------------
// out[row[e]] += scale * norm[e] * in[col[e]]   (per channel)

__global__ void edge_agg_kernel(float* __restrict__ out, const float* __restrict__ in,
                                const int* __restrict__ row, const int* __restrict__ col,
                                const float* __restrict__ norm, float scale, int total) {
    int idx = blockIdx.x * blockDim.x + threadIdx.x;
    if (idx >= total) return;
    int e = idx >> 7;
    int c = idx & (C - 1);
    float v = scale * norm[e] * in[col[e] * C + c];
    atomicAdd(&out[row[e] * C + c], v);
}

__global__ void neg_copy_kernel(float* __restrict__ dst, const float* __restrict__ src, int total) {
    int idx = blockIdx.x * blockDim.x + threadIdx.x;
    if (idx < total) dst[idx] = -src[idx];
}

// ---------------- dense ChebConv combine on WMMA f32 ---------------------
// Y = relu( X0@W[0] + X1@W[1] + X2@W[2] + b )
// block = 256 threads = 8 waves; block output stripe = 128 rows x 128 cols.
// Double-buffered LDS staging; weight chunk stored K-pair-interleaved so
// each B fragment is a single aligned ds_load_b64.

__launch_bounds__(256)
__global__ void cheb_dense_wmma(float* __restrict__ Y,
                                const float* __restrict__ X0,
                                const float* __restrict__ X1,
                                const float* __restrict__ X2,
                                const float* __restrict__ W,   // [3,128,128]
                                const float* __restrict__ bias,
                                int N, int applyRelu) {
    __shared__ float sIn[2 * C * 32];   // 2 x (128 rows x 32 K-cols)  (32 KB)
    __shared__ float sWp[2 * 32 * C];   // 2 x (16 K-pairs x 128 cols x 2)  (32 KB)

    const int tid  = threadIdx.x;
    const int wave = tid >> 5;      // 0..7
    const int lane = tid & 31;
    const int l16  = lane & 15;
    const int hi   = lane >> 4;     // 0 or 1 (selects K pair per ISA A/B layout)
    const int rowBase = blockIdx.x * 128;

    v8f acc[8] = {};                // 8 col tiles of 16x16 f32

    const float* Xs[3] = {X0, X1, X2};

    auto stage = [&](int s, int buf) {
        const float* X  = Xs[s >> 2];
        const float* Wm = W + (s >> 2) * C * C;
        const int kc = s & 3;
        float* bIn = sIn + buf * (C * 32);
        float* bW  = sWp + buf * (32 * C);
        // input rows (coalesced float4): 1024 float4
        for (int i = tid; i < 1024; i += 256) {
            int r  = i >> 3;
            int c4 = (i & 7) * 4;
            int gr = rowBase + r;
            float4 v = make_float4(0.f, 0.f, 0.f, 0.f);
            if (gr < N) v = *(const float4*)&X[gr * C + kc * 32 + c4];
            *(float4*)&bIn[r * 32 + c4] = v;
        }
        // weight chunk, K-pair interleaved: pair k2 holds rows {2k2, 2k2+1}
        for (int i = tid; i < 2048; i += 256) {
            int k2 = i >> 7;          // 0..15
            int c  = i & (C - 1);
            float2 v;
            v.x = Wm[(kc * 32 + 2 * k2 + 0) * C + c];
            v.y = Wm[(kc * 32 + 2 * k2 + 1) * C + c];
            *(float2*)&bW[(k2 * C + c) * 2] = v;
        }
    };

    auto compute = [&](int buf) {
        const float* bIn = sIn + buf * (C * 32);
        const float* bW  = sWp + buf * (32 * C);
        #pragma unroll
        for (int k4 = 0; k4 < 8; ++k4) {       // 8 K-steps of 4
            const int kk = k4 * 4 + hi * 2;    // A: lanes0-15 K{0,1}, lanes16-31 K{2,3}
            const int k2 = k4 * 2 + hi;        // matching interleaved B pair
            v2f a = *(const v2f*)&bIn[(wave * 16 + l16) * 32 + kk];
            #pragma unroll
            for (int ct = 0; ct < 8; ++ct) {
                v2f b = *(const v2f*)&bW[(k2 * C + ct * 16 + l16) * 2];
                acc[ct] = __builtin_amdgcn_wmma_f32_16x16x4_f32(
                    false, a, false, b, (short)0, acc[ct], false, false);
            }
        }
    };

    stage(0, 0);
    for (int s = 0; s < 12; ++s) {             // 3 matrices x 4 K-chunks
        __syncthreads();
        if (s + 1 < 12) stage(s + 1, (s + 1) & 1);
        compute(s & 1);
    }

    // epilogue: D layout — VGPR j: lanes0-15 -> M=j, lanes16-31 -> M=j+8
    #pragma unroll
    for (int ct = 0; ct < 8; ++ct) {
        #pragma unroll
        for (int j = 0; j < 8; ++j) {
            int gr = rowBase + wave * 16 + hi * 8 + j;
            int gc = ct * 16 + l16;
            if (gr < N) {
                float v = acc[ct][j] + bias[gc];
                if (applyRelu) v = fmaxf(v, 0.0f);
                Y[gr * C + gc] = v;
            }
        }
    }
}

// ---------------- batchnorm statistics -----------------------------------

__global__ void bn_partial_kernel(const float* __restrict__ h, float* __restrict__ sum,
                                  float* __restrict__ sumsq, int N) {
    __shared__ float ls[256], ls2[256];
    int c  = threadIdx.x & (C - 1);
    int rh = threadIdx.x >> 7;          // 0..1
    int base = blockIdx.x * 64;
    float s = 0.f, s2 = 0.f;
    for (int r = rh; r < 64; r += 2) {
        int n = base + r;
        if (n < N) {
            float v = h[n * C + c];
            s += v; s2 += v * v;
        }
    }
    ls[threadIdx.x] = s; ls2[threadIdx.x] = s2;
    __syncthreads();
    if (rh == 0) {
        atomicAdd(&sum[c],   ls[c]  + ls[c + 128]);
        atomicAdd(&sumsq[c], ls2[c] + ls2[c + 128]);
    }
}

__global__ void bn_final_kernel(const float* __restrict__ sum, const float* __restrict__ sumsq,
                                float* __restrict__ mu, float* __restrict__ rs, float n) {
    int c = threadIdx.x;
    float m = sum[c] / n;
    float v = sumsq[c] / n - m * m;
    mu[c] = m;
    rs[c] = rsqrtf(v + 1e-5f);
}

// ---------------- BN apply + mean pool ------------------------------------

__global__ void apply_pool_kernel(const float* __restrict__ h, const float* __restrict__ mu,
                                  const float* __restrict__ rs, const float* __restrict__ gamma,
                                  const float* __restrict__ beta, const int* __restrict__ batch,
                                  float* __restrict__ pooled, float* __restrict__ cnt, int N) {
    int idx = blockIdx.x * blockDim.x + threadIdx.x;
    if (idx >= N * C) return;
    int n = idx >> 7;
    int c = idx & (C - 1);
    float v = (h[idx] - mu[c]) * rs[c] * gamma[c] + beta[c];
    int g = batch[n];
    atomicAdd(&pooled[g * C + c], v);
    if (c == 0) atomicAdd(&cnt[g], 1.0f);
}

// ---------------- final MLP ------------------------------------------------

__global__ void mlp_kernel(const float* __restrict__ pooled, const float* __restrict__ cnt,
                           const float* __restrict__ lw1, const float* __restrict__ lb1,
                           const float* __restrict__ lw2, const float* __restrict__ lb2,
                           float* __restrict__ out) {
    __shared__ float p[C];
    __shared__ float t[16];
    int g = blockIdx.x;
    float cn = fmaxf(cnt[g], 1.0f);
    p[threadIdx.x] = pooled[g * C + threadIdx.x] / cn;
    __syncthreads();
    if (threadIdx.x < 16) {
        float a = lb1[threadIdx.x];
        for (int k = 0; k < C; ++k) a += p[k] * lw1[k * 16 + threadIdx.x];
        t[threadIdx.x] = fmaxf(a, 0.0f);
    }
    __syncthreads();
    if (threadIdx.x < 2) {
        float a = lb2[threadIdx.x];
        for (int k = 0; k < 16; ++k) a += t[k] * lw2[k * 2 + threadIdx.x];
        out[g * 2 + threadIdx.x] = a;
    }
}

// ---------------- host-side orchestration ---------------------------------

static inline size_t align256(size_t x) { return (x + 255) & ~size_t(255); }

extern "C" void kernel_launch(void* const* d_in, const int* in_sizes, int n_in,
                              void* d_out, int out_size, void* d_ws, size_t ws_size,
                              hipStream_t stream) {
    const float* atom_emb = (const float*)d_in[0];
    const float* W1    = (const float*)d_in[1];
    const float* b1    = (const float*)d_in[2];
    const float* W3    = (const float*)d_in[3];
    const float* b3    = (const float*)d_in[4];
    const float* gamma = (const float*)d_in[5];
    const float* beta  = (const float*)d_in[6];
    const float* lw1   = (const float*)d_in[7];
    const float* lb1   = (const float*)d_in[8];
    const float* lw2   = (const float*)d_in[9];
    const float* lb2   = (const float*)d_in[10];
    const int*   x     = (const int*)d_in[11];
    const int*   eidx  = (const int*)d_in[12];
    const int*   batch = (const int*)d_in[13];

    const int* row = eidx;
    const int* col = eidx + N_EDGES;

    // carve workspace
    char* ws = (char*)d_ws;
    size_t off = 0;
    float* deg   = (float*)(ws + off); off += align256(sizeof(float) * N_NODES);
    float* dis   = (float*)(ws + off); off += align256(sizeof(float) * N_NODES);
    float* norm  = (float*)(ws + off); off += align256(sizeof(float) * N_EDGES);
    float* hA    = (float*)(ws + off); off += align256(sizeof(float) * N_NODES * C);
    float* hB    = (float*)(ws + off); off += align256(sizeof(float) * N_NODES * C);
    float* Tx1   = (float*)(ws + off); off += align256(sizeof(float) * N_NODES * C);
    float* Tx2   = (float*)(ws + off); off += align256(sizeof(float) * N_NODES * C);
    float* sum   = (float*)(ws + off); off += align256(sizeof(float) * C);
    float* sumsq = (float*)(ws + off); off += align256(sizeof(float) * C);
    float* mu    = (float*)(ws + off); off += align256(sizeof(float) * C);
    float* rs    = (float*)(ws + off); off += align256(sizeof(float) * C);
    float* pooled= (float*)(ws + off); off += align256(sizeof(float) * N_GRAPHS * C);
    float* cnt   = (float*)(ws + off); off += align256(sizeof(float) * N_GRAPHS);
    (void)off; (void)ws_size; (void)in_sizes; (void)n_in; (void)out_size;

    const int NE_TOT = N_EDGES * C;
    const int NN_TOT = N_NODES * C;

    // 1) degree / normalization weights
    hipMemsetAsync(deg, 0, sizeof(float) * N_NODES, stream);
    deg_kernel<<<(N_EDGES + 255) / 256, 256, 0, stream>>>(row, deg, N_EDGES);
    dis_kernel<<<(N_NODES + 255) / 256, 256, 0, stream>>>(deg, dis, N_NODES);
    norm_kernel<<<(N_EDGES + 255) / 256, 256, 0, stream>>>(row, col, dis, norm, N_EDGES);

    // 2) atom encoder -> hA
    atom_encoder_kernel<<<N_NODES, C, 0, stream>>>(atom_emb, x, hA);

    const int denseGrid = (N_NODES + 127) / 128;

    // 3) conv1: hA -> hB
    hipMemsetAsync(Tx1, 0, sizeof(float) * NN_TOT, stream);
    edge_agg_kernel<<<(NE_TOT + 255) / 256, 256, 0, stream>>>(Tx1, hA, row, col, norm, 1.0f, NE_TOT);
    neg_copy_kernel<<<(NN_TOT + 255) / 256, 256, 0, stream>>>(Tx2, hA, NN_TOT);
    edge_agg_kernel<<<(NE_TOT + 255) / 256, 256, 0, stream>>>(Tx2, Tx1, row, col, norm, 2.0f, NE_TOT);
    cheb_dense_wmma<<<denseGrid, 256, 0, stream>>>(hB, hA, Tx1, Tx2, W1, b1, N_NODES, 1);

    // 4) conv3: hB -> hA
    hipMemsetAsync(Tx1, 0, sizeof(float) * NN_TOT, stream);
    edge_agg_kernel<<<(NE_TOT + 255) / 256, 256, 0, stream>>>(Tx1, hB, row, col, norm, 1.0f, NE_TOT);
    neg_copy_kernel<<<(NN_TOT + 255) / 256, 256, 0, stream>>>(Tx2, hB, NN_TOT);
    edge_agg_kernel<<<(NE_TOT + 255) / 256, 256, 0, stream>>>(Tx2, Tx1, row, col, norm, 2.0f, NE_TOT);
    cheb_dense_wmma<<<denseGrid, 256, 0, stream>>>(hA, hB, Tx1, Tx2, W3, b3, N_NODES, 1);

    // 5) batchnorm statistics
    hipMemsetAsync(sum, 0, sizeof(float) * C, stream);
    hipMemsetAsync(sumsq, 0, sizeof(float) * C, stream);
    bn_partial_kernel<<<(N_NODES + 63) / 64, 256, 0, stream>>>(hA, sum, sumsq, N_NODES);
    bn_final_kernel<<<1, C, 0, stream>>>(sum, sumsq, mu, rs, (float)N_NODES);

    // 6) BN apply + mean pool
    hipMemsetAsync(pooled, 0, sizeof(float) * N_GRAPHS * C, stream);
    hipMemsetAsync(cnt, 0, sizeof(float) * N_GRAPHS, stream);
    apply_pool_kernel<<<(NN_TOT + 255) / 256, 256, 0, stream>>>(hA, mu, rs, gamma, beta, batch,
                                                               pooled, cnt, N_NODES);

    // 7) MLP head
    mlp_kernel<<<N_GRAPHS, C, 0, stream>>>(pooled, cnt, lw1, lb1, lw2, lb2, (float*)d_out);
}